// ChamferDistanceLoss_11931419148400
// MI455X (gfx1250) — compile-verified
//
#include <hip/hip_runtime.h>
#include <hip/hip_bf16.h>

typedef float v2f __attribute__((ext_vector_type(2)));
typedef float v8f __attribute__((ext_vector_type(8)));

// Zero the scalar output accumulator (harness poisons d_out before timing).
__global__ void chamfer_init_kernel(float* out) {
    if (threadIdx.x == 0 && blockIdx.x == 0) out[0] = 0.0f;
}

// One wave32 per (batch, 16-row tile of Apts). Scans all tiles of Bpts.
// V_WMMA_F32_16X16X4_F32 with the embedding:
//   A row m = (-2*a0, -2*a1, -2*a2, 1)       (K=4)
//   B col n = ( b0,    b1,    b2,   |b|^2)
//   C       = 0   (inline constant -> no broadcast movs)
// => D[m][n] = |b_n|^2 - 2*a_m.b_n
// Since |a_m|^2 is constant over the min along n:
//   min_n dist(m,n) = |a_m|^2 + min_n D[m][n]
// and only SUM_m |a_m|^2 is needed at the end (cheap cross-lane sum).
__global__ __launch_bounds__(32) void chamfer_dir_kernel(
        const float* __restrict__ Apts,   // [B, N, 3] rows we take the min FOR
        const float* __restrict__ Bpts,   // [B, N, 3] points we take the min OVER
        float* __restrict__ out,          // scalar accumulator
        int N, float scale) {
    const int lane = threadIdx.x;          // 0..31 (wave32)
    const int idx  = lane & 15;
    const bool hi  = lane >= 16;
    const int tile  = blockIdx.x;          // which 16-row tile of Apts
    const int batch = blockIdx.y;

    const float* __restrict__ abase = Apts + (size_t)batch * (size_t)N * 3;
    const float* __restrict__ bbase = Bpts + (size_t)batch * (size_t)N * 3;

    // ---- Build A operand (held in 2 VGPRs for the whole scan) ----
    const float* arow = abase + (size_t)(tile * 16 + idx) * 3;
    const float pa0 = arow[0];
    const float pa1 = arow[1];
    const float pa2 = arow[2];
    const float psq = fmaf(pa0, pa0, fmaf(pa1, pa1, pa2 * pa2));

    // A 16x4 fp32 layout: VGPR0 = {K0 | K2}, VGPR1 = {K1 | K3} across lane halves
    v2f amat;
    amat.x = hi ? (-2.0f * pa2) : (-2.0f * pa0);   // K2 : K0
    amat.y = hi ?  1.0f         : (-2.0f * pa1);   // K3 : K1

    const float inf = __builtin_inff();
    float rm0 = inf, rm1 = inf, rm2 = inf, rm3 = inf;
    float rm4 = inf, rm5 = inf, rm6 = inf, rm7 = inf;

    const int ntiles = N >> 4;
    #pragma unroll 4
    for (int t = 0; t < ntiles; ++t) {
        const float* brow = bbase + (size_t)(t * 16 + idx) * 3;
        const float tb0 = brow[0];
        const float tb1 = brow[1];
        const float tb2 = brow[2];
        const float tsq = fmaf(tb0, tb0, fmaf(tb1, tb1, tb2 * tb2));

        // B 4x16 fp32 layout mirrors A: VGPR0 = {K0 | K2}, VGPR1 = {K1 | K3}
        v2f bmat;
        bmat.x = hi ? tb2 : tb0;     // K2 : K0
        bmat.y = hi ? tsq : tb1;     // K3 : K1

        v8f c = {};   // inline 0 accumulator

        // D = A*B + 0 = |b_n|^2 - 2*a_m.b_n
        v8f d = __builtin_amdgcn_wmma_f32_16x16x4_f32(
            /*neg_a=*/false, amat, /*neg_b=*/false, bmat,
            /*c_mod=*/(short)0, c, /*reuse_a=*/false, /*reuse_b=*/false);

        rm0 = fminf(rm0, d[0]);
        rm1 = fminf(rm1, d[1]);
        rm2 = fminf(rm2, d[2]);
        rm3 = fminf(rm3, d[3]);
        rm4 = fminf(rm4, d[4]);
        rm5 = fminf(rm5, d[5]);
        rm6 = fminf(rm6, d[6]);
        rm7 = fminf(rm7, d[7]);
    }

    // ---- Cross-lane reductions ----
    // rm[v] at lane L holds min over cols {n : n%16 == L%16} for row m = v + 8*(L>=16).
    // xor-reduce with masks 1,2,4,8 stays within each 16-lane half -> full row min.
    float rmv[8] = { rm0, rm1, rm2, rm3, rm4, rm5, rm6, rm7 };
    float total = 0.0f;
    #pragma unroll
    for (int v = 0; v < 8; ++v) {
        float r = rmv[v];
        r = fminf(r, __shfl_xor(r, 1, 32));
        r = fminf(r, __shfl_xor(r, 2, 32));
        r = fminf(r, __shfl_xor(r, 4, 32));
        r = fminf(r, __shfl_xor(r, 8, 32));
        total += r;   // half0 lanes: rows 0..7 ; half1 lanes: rows 8..15
    }
    total += __shfl_xor(total, 16, 32);   // merge halves: Sum_m min_n D[m][n]

    // Deferred |a_m|^2 term: Sum over the tile's 16 rows. Each row's psq is
    // duplicated in both halves, so the full 32-lane sum is 2x the tile sum.
    float ps = psq;
    ps += __shfl_xor(ps, 1, 32);
    ps += __shfl_xor(ps, 2, 32);
    ps += __shfl_xor(ps, 4, 32);
    ps += __shfl_xor(ps, 8, 32);
    ps += __shfl_xor(ps, 16, 32);
    total += 0.5f * ps;

    if (lane == 0) {
        atomicAdd(out, total * scale);
    }
}

extern "C" void kernel_launch(void* const* d_in, const int* in_sizes, int n_in,
                              void* d_out, int out_size, void* d_ws, size_t ws_size,
                              hipStream_t stream) {
    const float* pred   = (const float*)d_in[0];
    const float* target = (const float*)d_in[1];
    float* out = (float*)d_out;

    const int B = 8;
    const int N = in_sizes[0] / (B * 3);   // 4096
    const float scale = 1.0f / (float)(B * N);

    chamfer_init_kernel<<<1, 1, 0, stream>>>(out);

    dim3 grid(N / 16, B);
    // pred -> target direction (min over target for each pred row)
    chamfer_dir_kernel<<<grid, 32, 0, stream>>>(pred, target, out, N, scale);
    // target -> pred direction (construction is symmetric; swap roles)
    chamfer_dir_kernel<<<grid, 32, 0, stream>>>(target, pred, out, N, scale);
}